// Conv_764504178927
// MI455X (gfx1250) — compile-verified
//
#include <hip/hip_runtime.h>
#include <stdint.h>

// Problem geometry (fixed by reference): 4096x4096 fp32 in, 3x3 VALID conv, +bias
#define IH 4096
#define IW 4096
#define OH 4094
#define OW 4094

// Tiling: one workgroup (256 threads = 8 wave32) -> 128x32 output tile.
// Staged input tile: (128+2) x (32+2), padded to 136 floats/row so every
// async B128 copy is 16B aligned. LDS = 136*34*4 = 18,496 B.
#define TILE_W 128
#define TILE_H 32
#define LDS_W 136
#define LDS_H 34
#define F4_PER_ROW (LDS_W / 4)          // 34 float4 per staged row
#define N_F4 (LDS_H * F4_PER_ROW)       // 1156 async B128 copies per tile
#define NTHREADS 256

typedef __attribute__((address_space(3))) float lds_f;

__global__ __launch_bounds__(NTHREADS)
void conv3x3_async(const float* __restrict__ X,
                   const float* __restrict__ Wt,
                   const float* __restrict__ Bs,
                   float* __restrict__ Out)
{
    __shared__ float tile[LDS_H * LDS_W];

    const int tid = threadIdx.x;
    const int gx0 = blockIdx.x * TILE_W;   // global output col base of tile
    const int gy0 = blockIdx.y * TILE_H;   // global output row base of tile

    // LDS byte-offset base of the staged tile (addrspace(3) ptrtoint)
    lds_f* lb = (lds_f*)tile;

    // ---- Stage input tile: CDNA5 async global -> LDS, 16 B per lane ----
    // Edge tiles clamp gy/gx in-bounds; clamped (garbage) columns/rows are
    // only consumed by outputs that fall outside OHxOW and are never stored.
#pragma unroll
    for (int it = 0; it < (N_F4 + NTHREADS - 1) / NTHREADS; ++it) {
        int idx = tid + it * NTHREADS;
        if (idx < N_F4) {
            int row = idx / F4_PER_ROW;
            int c4  = idx - row * F4_PER_ROW;
            int gy  = gy0 + row;      if (gy > IH - 1) gy = IH - 1;
            int gx  = gx0 + c4 * 4;   if (gx > IW - 4) gx = IW - 4;
            const float* gp = X + (size_t)gy * IW + gx;        // 16B aligned
            unsigned loff = (unsigned)(uintptr_t)(lb + (row * LDS_W + c4 * 4));
            asm volatile("global_load_async_to_lds_b128 %0, %1, off"
                         :: "v"(loff), "v"(gp)
                         : "memory");
        }
    }
    // Wait for this wave's async copies, then make all waves' data visible.
    asm volatile("s_wait_asynccnt 0x0" ::: "memory");
    __syncthreads();

    // ---- Uniform weights + bias (scalar loads) ----
    const float w00 = Wt[0], w01 = Wt[1], w02 = Wt[2];
    const float w10 = Wt[3], w11 = Wt[4], w12 = Wt[5];
    const float w20 = Wt[6], w21 = Wt[7], w22 = Wt[8];
    const float bias = Bs[0];

    // ---- Each thread computes a 4x4 output micro-tile from a 6x6 window ----
    const int tx = tid & 31;        // 32 threads across x  -> cols 4*tx..4*tx+3
    const int ty = tid >> 5;        //  8 threads across y  -> rows 4*ty..4*ty+3
    const int br = ty * 4;          // tile-local row base (0..28)
    const int bc = tx * 4;          // tile-local col base (0..124), 16B aligned

    float a[6][6];
#pragma unroll
    for (int r = 0; r < 6; ++r) {
#pragma unroll
        for (int c = 0; c < 6; ++c)
            a[r][c] = tile[(br + r) * LDS_W + bc + c];   // ds_load b128/b64
    }

    float acc[4][4];
#pragma unroll
    for (int r = 0; r < 4; ++r) {
#pragma unroll
        for (int c = 0; c < 4; ++c) {
            float s = bias;
            s = fmaf(w00, a[r + 0][c + 0], s);
            s = fmaf(w01, a[r + 0][c + 1], s);
            s = fmaf(w02, a[r + 0][c + 2], s);
            s = fmaf(w10, a[r + 1][c + 0], s);
            s = fmaf(w11, a[r + 1][c + 1], s);
            s = fmaf(w12, a[r + 1][c + 2], s);
            s = fmaf(w20, a[r + 2][c + 0], s);
            s = fmaf(w21, a[r + 2][c + 1], s);
            s = fmaf(w22, a[r + 2][c + 2], s);
            acc[r][c] = s;
        }
    }

    // ---- Store: output row stride (4094 floats) is only 8B aligned, so use
    // float2 (b64) vector stores on the interior fast path. ----
    const int orow0 = gy0 + br;
    const int ocol0 = gx0 + bc;

    if (orow0 + 3 < OH && ocol0 + 3 < OW) {
#pragma unroll
        for (int r = 0; r < 4; ++r) {
            float2* p = (float2*)(Out + (size_t)(orow0 + r) * OW + ocol0);
            p[0] = make_float2(acc[r][0], acc[r][1]);
            p[1] = make_float2(acc[r][2], acc[r][3]);
        }
    } else {
#pragma unroll
        for (int r = 0; r < 4; ++r) {
            int orow = orow0 + r;
            if (orow < OH) {
#pragma unroll
                for (int c = 0; c < 4; ++c) {
                    int ocol = ocol0 + c;
                    if (ocol < OW)
                        Out[(size_t)orow * OW + ocol] = acc[r][c];
                }
            }
        }
    }
}

extern "C" void kernel_launch(void* const* d_in, const int* in_sizes, int n_in,
                              void* d_out, int out_size, void* d_ws, size_t ws_size,
                              hipStream_t stream)
{
    const float* X  = (const float*)d_in[0];   // (4096, 4096) fp32
    const float* Wt = (const float*)d_in[1];   // (3, 3) fp32
    const float* Bs = (const float*)d_in[2];   // (1,)  fp32
    float* Out = (float*)d_out;                // (4094, 4094) fp32

    dim3 grid((OW + TILE_W - 1) / TILE_W,      // 32
              (OH + TILE_H - 1) / TILE_H);     // 128
    conv3x3_async<<<grid, NTHREADS, 0, stream>>>(X, Wt, Bs, Out);
}